// ExplainerNC_66236985639226
// MI455X (gfx1250) — compile-verified
//
#include <hip/hip_runtime.h>
#include <hip/hip_bf16.h>

typedef __attribute__((ext_vector_type(2))) float v2f;
typedef __attribute__((ext_vector_type(8))) float v8f;

#define NN 8192
#define DD 128

// ---------------------------------------------------------------------------
// Zero a buffer with 128-bit stores (grid-stride).
// ---------------------------------------------------------------------------
__global__ void zero_f4_kernel(float4* __restrict__ p, long n4) {
    long idx = (long)blockIdx.x * blockDim.x + threadIdx.x;
    long stride = (long)gridDim.x * blockDim.x;
    float4 z = make_float4(0.f, 0.f, 0.f, 0.f);
    for (; idx < n4; idx += stride) p[idx] = z;
}

// ---------------------------------------------------------------------------
// Per-node dots via V_WMMA_F32_16X16X4_F32.
// One wave32 computes a 16(row)x16(col) f32 tile of embed[tile] @ Wp, where
// Wp[k][0] = W[k] (row-part), Wp[k][1] = W[D+k] (col-part), other cols 0.
// A (16x4 f32, 2 VGPR): lane l holds M=l%16, K = {0,1}+2*(l>=16)  (+k0)
// B (4x16 f32, 2 VGPR): lane l holds N=l%16, K = {0,1}+2*(l>=16)  (+k0)
// C (16x16 f32, 8 VGPR): acc[v] at lane l is M = v + 8*(l>=16), N = l%16
// B is built branchlessly: unconditional 64-bit loads of both W columns,
// blended with per-lane 0/1 masks so EXEC stays all-1s through the loop.
// ---------------------------------------------------------------------------
__global__ __launch_bounds__(32) void node_dots_wmma_kernel(
    const float* __restrict__ embed, const float* __restrict__ W,
    float* __restrict__ r, float* __restrict__ c) {
    const int lane = threadIdx.x;
    const int half = lane >> 4;      // 0 or 1
    const int lm   = lane & 15;      // 0..15
    const int rowBase = blockIdx.x * 16;

    const float m0 = (lm == 0) ? 1.0f : 0.0f;   // lane owns output col 0 (r)
    const float m1 = (lm == 1) ? 1.0f : 0.0f;   // lane owns output col 1 (c)

    const float2* __restrict__ arow =
        (const float2*)(embed + (size_t)(rowBase + lm) * DD) + half;  // pairs
    const float2* __restrict__ Wr = (const float2*)(W) + half;        // rows part
    const float2* __restrict__ Wc = (const float2*)(W + DD) + half;   // cols part

    v8f acc = {};
    #pragma unroll
    for (int t = 0; t < DD / 4; ++t) {
        const float2 av = arow[2 * t];
        const float2 wr = Wr[2 * t];
        const float2 wc = Wc[2 * t];
        v2f a; a.x = av.x; a.y = av.y;
        v2f b;
        b.x = m0 * wr.x + m1 * wc.x;
        b.y = m0 * wr.y + m1 * wc.y;
        acc = __builtin_amdgcn_wmma_f32_16x16x4_f32(
            /*neg_a=*/false, a, /*neg_b=*/false, b,
            /*c_mod=*/(short)0, acc, /*reuse_a=*/false, /*reuse_b=*/false);
    }
    if (lm < 2) {
        #pragma unroll
        for (int v = 0; v < 8; ++v) {
            const int node = rowBase + v + half * 8;
            if (lm == 0) r[node] = acc[v];
            else         c[node] = acc[v];
        }
    }
}

// ---------------------------------------------------------------------------
// s = dot(embed[nodeid], W[2D:3D]) + b   (one wave, shuffle reduction)
// ---------------------------------------------------------------------------
__global__ __launch_bounds__(32) void self_dot_kernel(
    const float* __restrict__ embed, const float* __restrict__ W,
    const float* __restrict__ b, const int* __restrict__ nodeid,
    float* __restrict__ s_out) {
    const int lane = threadIdx.x;
    const int nid = nodeid[0];
    float sum = 0.f;
    for (int k = lane; k < DD; k += 32)
        sum += embed[(size_t)nid * DD + k] * W[2 * DD + k];
    #pragma unroll
    for (int off = 16; off > 0; off >>= 1)
        sum += __shfl_down(sum, off, 32);
    if (lane == 0) s_out[0] = sum + b[0];
}

// ---------------------------------------------------------------------------
// Pass 1 over edges: concrete gate, scatter-add into dense mask M.
// ---------------------------------------------------------------------------
__global__ void edge_gate_kernel(
    const float* __restrict__ noise,
    const int* __restrict__ row, const int* __restrict__ col,
    const float* __restrict__ r, const float* __restrict__ c,
    const float* __restrict__ s, float* __restrict__ M, int E) {
    const int e = blockIdx.x * blockDim.x + threadIdx.x;
    if (e >= E) return;
    const int i = row[e];
    const int j = col[e];
    const float la = r[i] + c[j] + s[0];
    const float u = noise[e] * (1.0f - 2.0f * 1e-6f) + 1e-6f;
    const float z = (logf(u) - log1pf(-u) + la) * 2.0f;   // / BETA(0.5)
    const float gate = 1.0f / (1.0f + expf(-z));
    atomicAdd(&M[(size_t)i * NN + j], gate);
}

// ---------------------------------------------------------------------------
// Pass 2 over edges: out[i,j] += adj_data[e] * 0.5*(M[i,j] + M[j,i]), i != j.
// sym is identical for duplicate (i,j) edges, so folding it into the per-edge
// atomic contribution reproduces adj[i,j] * sym[i,j] exactly.
// ---------------------------------------------------------------------------
__global__ void edge_final_kernel(
    const float* __restrict__ adj_data,
    const int* __restrict__ row, const int* __restrict__ col,
    const float* __restrict__ M, float* __restrict__ out, int E) {
    const int e = blockIdx.x * blockDim.x + threadIdx.x;
    if (e >= E) return;
    const int i = row[e];
    const int j = col[e];
    if (i == j) return;
    const float sym = 0.5f * (M[(size_t)i * NN + j] + M[(size_t)j * NN + i]);
    atomicAdd(&out[(size_t)i * NN + j], adj_data[e] * sym);
}

extern "C" void kernel_launch(void* const* d_in, const int* in_sizes, int n_in,
                              void* d_out, int out_size, void* d_ws, size_t ws_size,
                              hipStream_t stream) {
    const float* embed    = (const float*)d_in[0];
    const float* adj_data = (const float*)d_in[1];
    const float* W        = (const float*)d_in[2];
    const float* b        = (const float*)d_in[3];
    const float* noise    = (const float*)d_in[4];
    const int*   adj_row  = (const int*)d_in[5];
    const int*   adj_col  = (const int*)d_in[6];
    const int*   nodeid   = (const int*)d_in[7];
    float* out = (float*)d_out;
    const int E = in_sizes[1];

    // Workspace layout: M (NN*NN f32), r (NN), c (NN), s (1)
    float* M = (float*)d_ws;
    float* r = M + (size_t)NN * NN;
    float* c = r + NN;
    float* s = c + NN;

    const long n4 = (long)NN * NN / 4;
    zero_f4_kernel<<<4096, 256, 0, stream>>>((float4*)out, n4);
    zero_f4_kernel<<<4096, 256, 0, stream>>>((float4*)M, n4);

    node_dots_wmma_kernel<<<NN / 16, 32, 0, stream>>>(embed, W, r, c);
    self_dot_kernel<<<1, 32, 0, stream>>>(embed, W, b, nodeid, s);

    const int blocks = (E + 255) / 256;
    edge_gate_kernel<<<blocks, 256, 0, stream>>>(noise, adj_row, adj_col, r, c, s, M, E);
    edge_final_kernel<<<blocks, 256, 0, stream>>>(adj_data, adj_row, adj_col, M, out, E);
}